// TopSample_90417651515415
// MI455X (gfx1250) — compile-verified
//
#include <hip/hip_runtime.h>
#include <hip/hip_bf16.h>

typedef __attribute__((ext_vector_type(2))) float v2f;
typedef __attribute__((ext_vector_type(8))) float v8f;

#define SEQ   8192
#define DIM   64
#define NHEAD 128   // B*H = 8*16
#define NC    512   // top-r threshold R

// ---------------------------------------------------------------------------
// Kernel 1: scores[head][j] = dot(q[head,0,:], k[head,j+1,:]) for j in [0,8191)
// One wave per 16-column tile using V_WMMA_F32_16X16X4_F32 (f32 precision).
// A (16x4): q row replicated across all 16 M-rows.
//   lane layout (ISA 7.12.2): half = lane>>4; a = { q[kb+2h], q[kb+2h+1] }
// B (4x16):  B[kk][n] = k[j0+n+1][kb+kk]
//   lane layout: n = lane&15;  b = { k_row[kb+2h], k_row[kb+2h+1] }
// C (16x16): row M=0 lives in c[0] of lanes 0..15 -> 16 scores per wave.
// ---------------------------------------------------------------------------
__global__ void __launch_bounds__(256)
topsample_score_kernel(const float* __restrict__ q,
                       const float* __restrict__ k,
                       float* __restrict__ scores)
{
    const int lane = threadIdx.x & 31;
    const int wave = threadIdx.x >> 5;
    const int t    = blockIdx.x * 8 + wave;   // global tile id
    const int head = t >> 9;                  // 512 tiles of 16 per head
    const int tile = t & 511;
    const int j0   = tile * 16;

    const float* q0 = q + (size_t)head * SEQ * DIM;         // q[b,h,0,:]
    const float* kh = k + (size_t)head * SEQ * DIM;

    const int half = lane >> 4;   // selects K pair {0,1} vs {2,3}
    const int ln   = lane & 15;   // N (column) for B, M (row) for A

    // k row feeding this lane's B fragment; clamp tail (j=8191 is discarded later)
    int row = j0 + ln + 1;
    if (row > SEQ - 1) row = SEQ - 1;
    const float* krow = kh + (size_t)row * DIM + 2 * half;
    const float* qrow = q0 + 2 * half;

    v8f c = {};
#pragma unroll
    for (int kb = 0; kb < DIM; kb += 4) {
        const float2 av = *(const float2*)(qrow + kb);
        const float2 bv = *(const float2*)(krow + kb);
        v2f a; a.x = av.x; a.y = av.y;
        v2f b; b.x = bv.x; b.y = bv.y;
        c = __builtin_amdgcn_wmma_f32_16x16x4_f32(
                /*neg_a=*/false, a, /*neg_b=*/false, b,
                /*c_mod=*/(short)0, c, /*reuse_a=*/false, /*reuse_b=*/false);
    }

    const int j = j0 + lane;                  // only lanes 0..15 hold row M=0
    if (lane < 16 && j < SEQ - 1)
        scores[(size_t)head * SEQ + j] = c[0];
}

// ---------------------------------------------------------------------------
// Kernel 2: zero the 4MB output (separate launch => ordered before scatter)
// ---------------------------------------------------------------------------
__global__ void __launch_bounds__(256)
topsample_zero_kernel(float4* __restrict__ out)
{
    out[(size_t)blockIdx.x * 256 + threadIdx.x] = make_float4(0.f, 0.f, 0.f, 0.f);
}

// ---------------------------------------------------------------------------
// Kernel 3: one block per head.
//  1) load 512 candidate scores, bitonic-sort values in LDS
//  2) histogram: each element binary-searches its lower-bound position
//  3) inclusive prefix H(t) = #{x <= c_t} = rank(c_t)+1  (distinct values)
//     scatter out[head][H(t)] = 1 ; out[head][0] = 1
// ---------------------------------------------------------------------------
__global__ void __launch_bounds__(256)
topsample_rank_kernel(const float* __restrict__ scores,
                      float* __restrict__ out)
{
    __shared__ float sv[NC];        // sorted candidate values
    __shared__ int   hist[NC + 1];  // insertion-position histogram

    const int tid  = threadIdx.x;
    const int head = blockIdx.x;
    const size_t base = (size_t)head * SEQ;

    // load candidates (scores j in [0,512) <=> original k rows 1..512)
    sv[tid]       = scores[base + tid];
    sv[tid + 256] = scores[base + tid + 256];
    hist[tid] = 0;
    hist[tid + 256] = 0;
    if (tid == 0) hist[NC] = 0;

    // bitonic sort of 512 values, 256 threads
    for (unsigned ksz = 2; ksz <= NC; ksz <<= 1) {
        for (unsigned jsz = ksz >> 1; jsz > 0; jsz >>= 1) {
            __syncthreads();
#pragma unroll
            for (unsigned i = tid; i < NC; i += 256) {
                unsigned p = i ^ jsz;
                if (p > i) {
                    bool up  = ((i & ksz) == 0);
                    float a = sv[i], b = sv[p];
                    if ((a > b) == up) { sv[i] = b; sv[p] = a; }
                }
            }
        }
    }
    __syncthreads();

    // histogram all 8191 valid elements by lower-bound position among sv[]
    for (int i = tid; i < SEQ - 1; i += 256) {
        const float x = scores[base + i];
        unsigned lo = 0, hi = NC;            // count of candidates < x
        while (lo < hi) {
            unsigned mid = (lo + hi) >> 1;
            lo = (sv[mid] < x) ? (mid + 1) : lo;
            hi = (sv[mid] < x) ? hi : mid;
        }
        atomicAdd(&hist[lo], 1);
    }
    __syncthreads();

    // serial inclusive scan + scatter (513 Trues per head). H(t) >= 1 and
    // strictly increasing (each candidate counts itself), max = 8191.
    if (tid == 0) {
        out[base + 0] = 1.0f;                // prepended zero index, always kept
        int run = 0;
#pragma unroll 4
        for (int t = 0; t < NC; ++t) {
            run += hist[t];                  // rank(c_t) + 1
            out[base + run] = 1.0f;
        }
    }
}

extern "C" void kernel_launch(void* const* d_in, const int* in_sizes, int n_in,
                              void* d_out, int out_size, void* d_ws, size_t ws_size,
                              hipStream_t stream)
{
    const float* q = (const float*)d_in[0];  // [8,16,8192,64] f32
    const float* k = (const float*)d_in[1];  // [8,16,8192,64] f32
    float* out     = (float*)d_out;          // [8,16,8192] mask as f32
    float* scores  = (float*)d_ws;           // 128*8192 f32 = 4 MB scratch

    // 128 heads * 512 tiles = 65536 waves; 8 waves / block
    topsample_score_kernel<<<8192, 256, 0, stream>>>(q, k, scores);
    // 1024 * 256 * float4 = 1,048,576 floats
    topsample_zero_kernel<<<1024, 256, 0, stream>>>((float4*)d_out);
    // one block per head
    topsample_rank_kernel<<<NHEAD, 256, 0, stream>>>(scores, out);
}